// GraphPreProcessingLayer_27230092656694
// MI455X (gfx1250) — compile-verified
//
#include <hip/hip_runtime.h>
#include <math.h>

#define N 8192
#define TILE 64
#define TB (N / TILE)        // 128 tile-blocks per side
#define LDS_STRIDE 68        // 64 + 4 pad: 16B-aligned rows, conflict-free transposed reads

typedef __attribute__((ext_vector_type(2))) float v2f;
typedef __attribute__((ext_vector_type(8))) float v8f;

// Load one 64x64 fp32 tile (rowBlk, colBlk) of adj into LDS, coalesced b128.
__device__ __forceinline__ void load_tile(const float* __restrict__ adj,
                                          float (*lds)[LDS_STRIDE],
                                          int rowBlk, int colBlk, int tid) {
  // 64 rows x 16 float4 = 1024 float4; 256 threads -> 4 each
#pragma unroll
  for (int k = 0; k < 4; ++k) {
    const int idx = tid + k * 256;
    const int r  = idx >> 4;
    const int c4 = (idx & 15) << 2;
    const float4 v = *reinterpret_cast<const float4*>(
        adj + (size_t)(rowBlk * TILE + r) * N + (size_t)colBlk * TILE + c4);
    *reinterpret_cast<float4*>(&lds[r][c4]) = v;
  }
}

// Pass 1: rowsum[i] += sum_j max(A[i][j], A[j][i]) via triangular tiles.
// Row sums computed with V_WMMA_F32_16X16X4_F32 against an all-ones B (exact fp32).
__global__ void __launch_bounds__(256)
gcn_rowsum_kernel(const float* __restrict__ adj, float* __restrict__ rowsum) {
  __shared__ float ldsA[TILE][LDS_STRIDE];
  __shared__ float ldsB[TILE][LDS_STRIDE];
  const int jB = blockIdx.x, iB = blockIdx.y;
  if (iB > jB) return;  // triangular: each element of adj read exactly once
  const int tid = threadIdx.x;

  load_tile(adj, ldsA, iB, jB, tid);  // A tile  [iB, jB]
  load_tile(adj, ldsB, jB, iB, tid);  // AT tile [jB, iB]
  __syncthreads();

  const int wave = tid >> 5;
  const int lane = tid & 31;
  const bool colside = (wave >= 4);       // waves 4..7: column sums of M
  if (colside && iB == jB) return;        // diagonal tile: count each element once
  const int stripe = colside ? (wave - 4) : wave;  // 16-row stripe within tile
  // M[r][c] = max(P[r][c], Q[c][r]); colside swaps P/Q => operates on M^T
  float (*P)[LDS_STRIDE] = colside ? ldsB : ldsA;
  float (*Q)[LDS_STRIDE] = colside ? ldsA : ldsB;
  const int rowsBase = (colside ? jB : iB) * TILE + stripe * 16;

  // A-matrix 16x4 f32 layout: lane L -> row L%16, K pair at 2*(L/16)
  const int r     = stripe * 16 + (lane & 15);
  const int khalf = (lane >> 4) << 1;  // 0 or 2
  v2f ones; ones[0] = 1.0f; ones[1] = 1.0f;
  v8f acc = {0.f, 0.f, 0.f, 0.f, 0.f, 0.f, 0.f, 0.f};
#pragma unroll
  for (int k = 0; k < TILE; k += 4) {
    const int c0 = k + khalf;
    v2f a;
    a[0] = fmaxf(P[r][c0],     Q[c0][r]);
    a[1] = fmaxf(P[r][c0 + 1], Q[c0 + 1][r]);
    // D = A(16x4) x ones(4x16) + C : 16 exact fp32 row sums per stripe
    acc = __builtin_amdgcn_wmma_f32_16x16x4_f32(false, a, false, ones,
                                                (short)0, acc, false, false);
  }

  // C/D layout: VGPR v -> row v (lanes 0-15) / row v+8 (lanes 16-31), replicated over N.
  if ((lane & 15) == 0) {
    float* dst = rowsum + rowsBase + ((lane >> 4) << 3);
    atomicAdd(dst + 0, acc[0]);
    atomicAdd(dst + 1, acc[1]);
    atomicAdd(dst + 2, acc[2]);
    atomicAdd(dst + 3, acc[3]);
    atomicAdd(dst + 4, acc[4]);
    atomicAdd(dst + 5, acc[5]);
    atomicAdd(dst + 6, acc[6]);
    atomicAdd(dst + 7, acc[7]);
  }
}

// Pass 2: dinv[i] = (rowsum[i] + 1)^-1/2  (self-loop), inf -> 0 like reference.
__global__ void gcn_dinv_kernel(const float* __restrict__ rowsum,
                                float* __restrict__ dinv) {
  const int i = blockIdx.x * blockDim.x + threadIdx.x;
  if (i < N) {
    const float s = rowsum[i] + 1.0f;
    float d = rsqrtf(s);
    if (isinf(d)) d = 0.0f;
    dinv[i] = d;
  }
}

// Pass 3: recompute M per tile, scale, write both mirrored tiles.
__global__ void __launch_bounds__(256)
gcn_scale_kernel(const float* __restrict__ adj, const float* __restrict__ dinv,
                 float* __restrict__ out) {
  __shared__ float ldsA[TILE][LDS_STRIDE];
  __shared__ float ldsB[TILE][LDS_STRIDE];
  const int jB = blockIdx.x, iB = blockIdx.y;
  if (iB > jB) return;
  const int tid = threadIdx.x;

  load_tile(adj, ldsA, iB, jB, tid);
  load_tile(adj, ldsB, jB, iB, tid);
  __syncthreads();

#pragma unroll
  for (int k = 0; k < 4; ++k) {
    const int idx = tid + k * 256;
    const int r  = idx >> 4;
    const int c4 = (idx & 15) << 2;
    const float di = dinv[iB * TILE + r];
    float4 o;
    float* op = &o.x;
#pragma unroll
    for (int t = 0; t < 4; ++t) {
      const int c = c4 + t;
      float mv = fmaxf(ldsA[r][c], ldsB[c][r]);
      if (iB == jB && r == c) mv += 1.0f;  // self-loop on global diagonal
      op[t] = di * mv * dinv[jB * TILE + c];
    }
    *reinterpret_cast<float4*>(out + (size_t)(iB * TILE + r) * N +
                               (size_t)jB * TILE + c4) = o;
  }
  if (iB != jB) {  // mirrored tile, coalesced via the transposed LDS view
#pragma unroll
    for (int k = 0; k < 4; ++k) {
      const int idx = tid + k * 256;
      const int r  = idx >> 4;
      const int c4 = (idx & 15) << 2;
      const float di = dinv[jB * TILE + r];
      float4 o;
      float* op = &o.x;
#pragma unroll
      for (int t = 0; t < 4; ++t) {
        const int c = c4 + t;
        const float mv = fmaxf(ldsB[r][c], ldsA[c][r]);
        op[t] = di * mv * dinv[iB * TILE + c];
      }
      *reinterpret_cast<float4*>(out + (size_t)(jB * TILE + r) * N +
                                 (size_t)iB * TILE + c4) = o;
    }
  }
}

extern "C" void kernel_launch(void* const* d_in, const int* in_sizes, int n_in,
                              void* d_out, int out_size, void* d_ws, size_t ws_size,
                              hipStream_t stream) {
  (void)in_sizes; (void)n_in; (void)out_size; (void)ws_size;
  const float* adj = (const float*)d_in[0];
  float* out    = (float*)d_out;
  float* rowsum = (float*)d_ws;       // N floats
  float* dinv   = rowsum + N;         // N floats

  hipMemsetAsync(rowsum, 0, N * sizeof(float), stream);  // ws is poisoned; must zero

  dim3 grid(TB, TB), block(256);
  gcn_rowsum_kernel<<<grid, block, 0, stream>>>(adj, rowsum);
  gcn_dinv_kernel<<<(N + 255) / 256, 256, 0, stream>>>(rowsum, dinv);
  gcn_scale_kernel<<<grid, block, 0, stream>>>(adj, dinv, out);
}